// ViT_26121991094869
// MI455X (gfx1250) — compile-verified
//
#include <hip/hip_runtime.h>
#include <math.h>

#define B_  512
#define S_  100
#define L_  101
#define T_  (B_ * L_)          // 51712 = 16 * 3232
#define DE  150                // embed half
#define DZ  300                // token dim
#define DM  600                // msa dim
#define SCP 104                // padded attn row stride (101 -> 104, mult of 4)
#define EPSF 1e-5

typedef __attribute__((ext_vector_type(2))) float v2f;
typedef __attribute__((ext_vector_type(8))) float v8f;

#define WMMA_F32(acc, a, b) \
  (acc) = __builtin_amdgcn_wmma_f32_16x16x4_f32(false, (a), false, (b), (short)0, (acc), false, false)

// ---------------------------------------------------------------- fill z (cls + pos)
__global__ void k_fill_z(float* __restrict__ z, const float* __restrict__ pos,
                         const float* __restrict__ cls) {
  long idx = (long)blockIdx.x * blockDim.x + threadIdx.x;
  const long N1 = (long)B_ * L_ * DE;          // pos halves for all rows
  const long N2 = (long)B_ * DE;               // cls half for row 0
  if (idx < N1) {
    int b = (int)(idx / (L_ * DE));
    int r = (int)(idx % (L_ * DE));
    int l = r / DE, f = r % DE;
    z[((long)(b * L_ + l)) * DZ + DE + f] = pos[l * DE + f];
  } else if (idx < N1 + N2) {
    long j = idx - N1;
    int b = (int)(j / DE), f = (int)(j % DE);
    z[((long)(b * L_)) * DZ + f] = cls[f];
  }
}

// ---------------------------------------------------------------- patch embed: relu(x@Wp+bp) -> z[:, :150]
// grid(3, 3200): 12 N-tiles (cols 0..191, clamped loads, guarded stores)
__global__ void k_gemm_patch(const float* __restrict__ x, const float* __restrict__ Wp,
                             const float* __restrict__ bp, float* __restrict__ z) {
  const int lane = threadIdx.x & 31;
  const int n16 = lane & 15, half = lane >> 4;
  const int m0 = blockIdx.y * 16;
  int n[4], nc[4];
  #pragma unroll
  for (int j = 0; j < 4; ++j) {
    n[j] = blockIdx.x * 64 + j * 16 + n16;
    nc[j] = n[j] < DE ? n[j] : (DE - 1);       // clamp: garbage cols never stored
  }
  const float* arow = x + (long)(m0 + n16) * DZ;
  v8f acc[4] = {{0.f,0.f,0.f,0.f,0.f,0.f,0.f,0.f}, {0.f,0.f,0.f,0.f,0.f,0.f,0.f,0.f},
                {0.f,0.f,0.f,0.f,0.f,0.f,0.f,0.f}, {0.f,0.f,0.f,0.f,0.f,0.f,0.f,0.f}};
  for (int k0 = 0; k0 < DZ; k0 += 4) {
    int ka = k0 + 2 * half;
    v2f a = *(const v2f*)(arow + ka);
    v2f bf[4];
    #pragma unroll
    for (int j = 0; j < 4; ++j) {
      bf[j].x = Wp[(long)ka * DE + nc[j]];
      bf[j].y = Wp[(long)(ka + 1) * DE + nc[j]];
    }
    #pragma unroll
    for (int j = 0; j < 4; ++j) WMMA_F32(acc[j], a, bf[j]);
  }
  #pragma unroll
  for (int j = 0; j < 4; ++j) {
    if (n[j] < DE) {
      float bias = bp[n[j]];
      #pragma unroll
      for (int r = 0; r < 8; ++r) {
        int t = m0 + r + 8 * half;               // 0..51199
        int bb = t / S_, s = t % S_;
        float v = acc[j][r] + bias;
        z[((long)(bb * L_ + s + 1)) * DZ + n[j]] = v > 0.f ? v : 0.f;
      }
    }
  }
}

// ---------------------------------------------------------------- z @ W[300x300] -> out (Q/K/V per head)
// grid(5, 3232): 20 N-tiles (cols 0..319 clamped)
__global__ void k_gemm_proj(const float* __restrict__ z, const float* __restrict__ W,
                            float* __restrict__ out) {
  const int lane = threadIdx.x & 31;
  const int n16 = lane & 15, half = lane >> 4;
  const int m0 = blockIdx.y * 16;
  int n[4], nc[4];
  #pragma unroll
  for (int j = 0; j < 4; ++j) {
    n[j] = blockIdx.x * 64 + j * 16 + n16;
    nc[j] = n[j] < DZ ? n[j] : (DZ - 1);
  }
  const float* arow = z + (long)(m0 + n16) * DZ;
  v8f acc[4] = {{0.f,0.f,0.f,0.f,0.f,0.f,0.f,0.f}, {0.f,0.f,0.f,0.f,0.f,0.f,0.f,0.f},
                {0.f,0.f,0.f,0.f,0.f,0.f,0.f,0.f}, {0.f,0.f,0.f,0.f,0.f,0.f,0.f,0.f}};
  for (int k0 = 0; k0 < DZ; k0 += 4) {
    int ka = k0 + 2 * half;
    v2f a = *(const v2f*)(arow + ka);
    v2f bf[4];
    #pragma unroll
    for (int j = 0; j < 4; ++j) {
      bf[j].x = W[(long)ka * DZ + nc[j]];
      bf[j].y = W[(long)(ka + 1) * DZ + nc[j]];
    }
    #pragma unroll
    for (int j = 0; j < 4; ++j) WMMA_F32(acc[j], a, bf[j]);
  }
  #pragma unroll
  for (int j = 0; j < 4; ++j) {
    if (n[j] < DZ) {
      #pragma unroll
      for (int r = 0; r < 8; ++r)
        out[(long)(m0 + r + 8 * half) * DZ + n[j]] = acc[j][r];
    }
  }
}

// ---------------------------------------------------------------- scores = Q @ K^T per (h,b)
// grid(2, 7, 1024): 8 N-tiles (rows of K up to 127, overread stays in-ws)
__global__ void k_gemm_scores(const float* __restrict__ Q, const float* __restrict__ Km,
                              float* __restrict__ sc) {
  const int hb = blockIdx.z;                   // h*512 + b
  const int h = hb >> 9, b = hb & 511;
  const float* Qb = Q  + ((long)h * T_ + (long)b * L_) * DZ;
  const float* Kb = Km + ((long)h * T_ + (long)b * L_) * DZ;
  float* scb = sc + (long)hb * L_ * SCP;
  const int lane = threadIdx.x & 31;
  const int n16 = lane & 15, half = lane >> 4;
  const int m0 = blockIdx.y * 16;
  int n[4];
  #pragma unroll
  for (int j = 0; j < 4; ++j) n[j] = blockIdx.x * 64 + j * 16 + n16;
  const float* arow = Qb + (long)(m0 + n16) * DZ;
  v8f acc[4] = {{0.f,0.f,0.f,0.f,0.f,0.f,0.f,0.f}, {0.f,0.f,0.f,0.f,0.f,0.f,0.f,0.f},
                {0.f,0.f,0.f,0.f,0.f,0.f,0.f,0.f}, {0.f,0.f,0.f,0.f,0.f,0.f,0.f,0.f}};
  for (int k0 = 0; k0 < DZ; k0 += 4) {
    int ka = k0 + 2 * half;
    v2f a = *(const v2f*)(arow + ka);
    v2f bf[4];
    #pragma unroll
    for (int j = 0; j < 4; ++j)
      bf[j] = *(const v2f*)(Kb + (long)n[j] * DZ + ka);   // K^T: contiguous in K's row n
    #pragma unroll
    for (int j = 0; j < 4; ++j) WMMA_F32(acc[j], a, bf[j]);
  }
  #pragma unroll
  for (int j = 0; j < 4; ++j) {
    if (n[j] < L_) {
      #pragma unroll
      for (int r = 0; r < 8; ++r) {
        int m = m0 + r + 8 * half;
        if (m < L_) scb[(long)m * SCP + n[j]] = acc[j][r];
      }
    }
  }
}

// ---------------------------------------------------------------- row softmax (one wave32 per row), zero pad cols
__global__ void k_softmax(float* __restrict__ sc) {
  float* p = sc + (long)blockIdx.x * SCP;
  const int lane = threadIdx.x & 31;
  float vals[4];
  float mx = -3.4e38f;
  #pragma unroll
  for (int i = 0; i < 4; ++i) {
    int c = lane + 32 * i;
    float v = (c < L_) ? p[c] : -3.4e38f;
    vals[i] = v; mx = fmaxf(mx, v);
  }
  for (int o = 16; o > 0; o >>= 1) mx = fmaxf(mx, __shfl_xor(mx, o, 32));
  float s = 0.f;
  #pragma unroll
  for (int i = 0; i < 4; ++i) {
    int c = lane + 32 * i;
    float e = (c < L_) ? expf(vals[i] - mx) : 0.f;
    vals[i] = e; s += e;
  }
  for (int o = 16; o > 0; o >>= 1) s += __shfl_xor(s, o, 32);
  float inv = 1.f / s;
  #pragma unroll
  for (int i = 0; i < 4; ++i) {
    int c = lane + 32 * i;
    if (c < SCP) p[c] = (c < L_) ? vals[i] * inv : 0.f;   // zero K-pad cols 101..103
  }
}

// ---------------------------------------------------------------- msa[:, h*300:] = attn @ V per (h,b)
// grid(5, 7, 1024): 20 N-tiles; V col overreads stay in row+1 (in-ws)
__global__ void k_gemm_attnv(const float* __restrict__ sc, const float* __restrict__ V,
                             float* __restrict__ msa) {
  const int hb = blockIdx.z;
  const int h = hb >> 9, b = hb & 511;
  const float* Ab = sc + (long)hb * L_ * SCP;
  const float* Vb = V + ((long)h * T_ + (long)b * L_) * DZ;
  const int lane = threadIdx.x & 31;
  const int n16 = lane & 15, half = lane >> 4;
  const int m0 = blockIdx.y * 16;
  int n[4];
  #pragma unroll
  for (int j = 0; j < 4; ++j) n[j] = blockIdx.x * 64 + j * 16 + n16;
  const float* arow = Ab + (long)(m0 + n16) * SCP;
  v8f acc[4] = {{0.f,0.f,0.f,0.f,0.f,0.f,0.f,0.f}, {0.f,0.f,0.f,0.f,0.f,0.f,0.f,0.f},
                {0.f,0.f,0.f,0.f,0.f,0.f,0.f,0.f}, {0.f,0.f,0.f,0.f,0.f,0.f,0.f,0.f}};
  for (int k0 = 0; k0 < SCP; k0 += 4) {      // attn cols 101..103 are exact zeros
    int ka = k0 + 2 * half;
    v2f a = *(const v2f*)(arow + ka);
    v2f bf[4];
    #pragma unroll
    for (int j = 0; j < 4; ++j) {
      bf[j].x = Vb[(long)ka * DZ + n[j]];
      bf[j].y = Vb[(long)(ka + 1) * DZ + n[j]];
    }
    #pragma unroll
    for (int j = 0; j < 4; ++j) WMMA_F32(acc[j], a, bf[j]);
  }
  #pragma unroll
  for (int j = 0; j < 4; ++j) {
    if (n[j] < DZ) {
      #pragma unroll
      for (int r = 0; r < 8; ++r) {
        int m = m0 + r + 8 * half;
        if (m < L_) msa[((long)(b * L_ + m)) * DM + (long)h * DZ + n[j]] = acc[j][r];
      }
    }
  }
}

// ---------------------------------------------------------------- BN stats: per channel (token pos) over (B, F)
__global__ void k_bn_stats(const float* __restrict__ buf, int F,
                           float* __restrict__ mean, float* __restrict__ rstd) {
  const int ch = blockIdx.x;
  const int tid = threadIdx.x;
  const long cnt = (long)B_ * F;
  double s = 0.0, q = 0.0;
  for (long i = tid; i < cnt; i += blockDim.x) {
    long b = i / F, f = i % F;
    float v = buf[((long)(b * L_ + ch)) * F + f];
    s += v; q += (double)v * v;
  }
  __shared__ double sh_s[256], sh_q[256];
  sh_s[tid] = s; sh_q[tid] = q; __syncthreads();
  for (int o = 128; o > 0; o >>= 1) {
    if (tid < o) { sh_s[tid] += sh_s[tid + o]; sh_q[tid] += sh_q[tid + o]; }
    __syncthreads();
  }
  if (tid == 0) {
    double m = sh_s[0] / (double)cnt;
    double var = sh_q[0] / (double)cnt - m * m;
    mean[ch] = (float)m;
    rstd[ch] = (float)(1.0 / sqrt(var + EPSF));
  }
}

__global__ void k_bn_apply(float* __restrict__ buf, int F, const float* __restrict__ mean,
                           const float* __restrict__ rstd, const float* __restrict__ g,
                           const float* __restrict__ bta, long total) {
  long idx = (long)blockIdx.x * blockDim.x + threadIdx.x;
  if (idx >= total) return;
  long per = (long)L_ * F;
  int ch = (int)((idx % per) / F);
  float v = buf[idx];
  buf[idx] = (v - mean[ch]) * rstd[ch] * g[ch] + bta[ch];
}

// ---------------------------------------------------------------- y1 = relu(concat(bn1(msa), z) @ Wp1 + bp1)
// grid(5, 3232); K split 600 (msa) + 300 (z), both regions mult-of-4 aligned
__global__ void k_gemm_p1(const float* __restrict__ msa, const float* __restrict__ z,
                          const float* __restrict__ W, const float* __restrict__ bias,
                          float* __restrict__ y1) {
  const int lane = threadIdx.x & 31;
  const int n16 = lane & 15, half = lane >> 4;
  const int m0 = blockIdx.y * 16;
  int n[4], nc[4];
  #pragma unroll
  for (int j = 0; j < 4; ++j) {
    n[j] = blockIdx.x * 64 + j * 16 + n16;
    nc[j] = n[j] < DZ ? n[j] : (DZ - 1);
  }
  const long arow = m0 + n16;
  const float* Am = msa + arow * DM;
  const float* Az = z   + arow * DZ;
  v8f acc[4] = {{0.f,0.f,0.f,0.f,0.f,0.f,0.f,0.f}, {0.f,0.f,0.f,0.f,0.f,0.f,0.f,0.f},
                {0.f,0.f,0.f,0.f,0.f,0.f,0.f,0.f}, {0.f,0.f,0.f,0.f,0.f,0.f,0.f,0.f}};
  for (int k0 = 0; k0 < DM; k0 += 4) {
    int ka = k0 + 2 * half;
    v2f a = *(const v2f*)(Am + ka);
    v2f bf[4];
    #pragma unroll
    for (int j = 0; j < 4; ++j) {
      bf[j].x = W[(long)ka * DZ + nc[j]];
      bf[j].y = W[(long)(ka + 1) * DZ + nc[j]];
    }
    #pragma unroll
    for (int j = 0; j < 4; ++j) WMMA_F32(acc[j], a, bf[j]);
  }
  for (int k0 = 0; k0 < DZ; k0 += 4) {
    int ka = k0 + 2 * half;
    v2f a = *(const v2f*)(Az + ka);
    v2f bf[4];
    #pragma unroll
    for (int j = 0; j < 4; ++j) {
      bf[j].x = W[(long)(DM + ka) * DZ + nc[j]];
      bf[j].y = W[(long)(DM + ka + 1) * DZ + nc[j]];
    }
    #pragma unroll
    for (int j = 0; j < 4; ++j) WMMA_F32(acc[j], a, bf[j]);
  }
  #pragma unroll
  for (int j = 0; j < 4; ++j) {
    if (n[j] < DZ) {
      float bi = bias[n[j]];
      #pragma unroll
      for (int r = 0; r < 8; ++r) {
        float v = acc[j][r] + bi;
        y1[(long)(m0 + r + 8 * half) * DZ + n[j]] = v > 0.f ? v : 0.f;
      }
    }
  }
}

// ---------------------------------------------------------------- hid = concat(bn2(y1), bn1(msa), z)[:,0,:] @ fc1_W + fc1_b
// grid(8, 32); N=512 exact, K split 300 + 600 + 300
__global__ void k_gemm_fc1(const float* __restrict__ y1, const float* __restrict__ msa,
                           const float* __restrict__ z, const float* __restrict__ W,
                           const float* __restrict__ bias, float* __restrict__ hid) {
  const int lane = threadIdx.x & 31;
  const int n16 = lane & 15, half = lane >> 4;
  const int m0 = blockIdx.y * 16;
  int n[4];
  #pragma unroll
  for (int j = 0; j < 4; ++j) n[j] = blockIdx.x * 64 + j * 16 + n16;
  const long brow = m0 + n16;                  // batch index, token 0 rows
  const float* Ay = y1  + brow * (long)L_ * DZ;
  const float* Am = msa + brow * (long)L_ * DM;
  const float* Az = z   + brow * (long)L_ * DZ;
  v8f acc[4] = {{0.f,0.f,0.f,0.f,0.f,0.f,0.f,0.f}, {0.f,0.f,0.f,0.f,0.f,0.f,0.f,0.f},
                {0.f,0.f,0.f,0.f,0.f,0.f,0.f,0.f}, {0.f,0.f,0.f,0.f,0.f,0.f,0.f,0.f}};
  #pragma unroll 1
  for (int k0 = 0; k0 < 300; k0 += 4) {
    int ka = k0 + 2 * half;
    v2f a = *(const v2f*)(Ay + ka);
    v2f bf[4];
    #pragma unroll
    for (int j = 0; j < 4; ++j) {
      bf[j].x = W[(long)ka * 512 + n[j]];
      bf[j].y = W[(long)(ka + 1) * 512 + n[j]];
    }
    #pragma unroll
    for (int j = 0; j < 4; ++j) WMMA_F32(acc[j], a, bf[j]);
  }
  #pragma unroll 1
  for (int k0 = 0; k0 < 600; k0 += 4) {
    int ka = k0 + 2 * half;
    v2f a = *(const v2f*)(Am + ka);
    v2f bf[4];
    #pragma unroll
    for (int j = 0; j < 4; ++j) {
      bf[j].x = W[(long)(300 + ka) * 512 + n[j]];
      bf[j].y = W[(long)(300 + ka + 1) * 512 + n[j]];
    }
    #pragma unroll
    for (int j = 0; j < 4; ++j) WMMA_F32(acc[j], a, bf[j]);
  }
  #pragma unroll 1
  for (int k0 = 0; k0 < 300; k0 += 4) {
    int ka = k0 + 2 * half;
    v2f a = *(const v2f*)(Az + ka);
    v2f bf[4];
    #pragma unroll
    for (int j = 0; j < 4; ++j) {
      bf[j].x = W[(long)(900 + ka) * 512 + n[j]];
      bf[j].y = W[(long)(900 + ka + 1) * 512 + n[j]];
    }
    #pragma unroll
    for (int j = 0; j < 4; ++j) WMMA_F32(acc[j], a, bf[j]);
  }
  #pragma unroll
  for (int j = 0; j < 4; ++j) {
    #pragma unroll
    for (int r = 0; r < 8; ++r)
      hid[(long)(m0 + r + 8 * half) * 512 + n[j]] = acc[j][r] + bias[n[j]];
  }
}

// ---------------------------------------------------------------- out = hid @ fc2_W + fc2_b  (tiny)
__global__ void k_fc2(const float* __restrict__ hid, const float* __restrict__ W,
                      const float* __restrict__ bias, float* __restrict__ out) {
  int t = blockIdx.x * blockDim.x + threadIdx.x;
  if (t >= B_ * 2) return;
  int b = t >> 1, j = t & 1;
  const float* h = hid + (long)b * 512;
  float s = bias[j];
  for (int k = 0; k < 512; ++k) s = fmaf(h[k], W[k * 2 + j], s);
  out[t] = s;
}

// ================================================================ launch
extern "C" void kernel_launch(void* const* d_in, const int* in_sizes, int n_in,
                              void* d_out, int out_size, void* d_ws, size_t ws_size,
                              hipStream_t stream) {
  const float* x    = (const float*)d_in[0];
  const float* Wp   = (const float*)d_in[1];
  const float* bp   = (const float*)d_in[2];
  const float* pos  = (const float*)d_in[3];
  const float* cls  = (const float*)d_in[4];
  const float* Wq   = (const float*)d_in[5];
  const float* Wk   = (const float*)d_in[6];
  const float* Wv   = (const float*)d_in[7];
  const float* bn1g = (const float*)d_in[8];
  const float* bn1b = (const float*)d_in[9];
  const float* Wp1  = (const float*)d_in[10];
  const float* bp1  = (const float*)d_in[11];
  const float* bn2g = (const float*)d_in[12];
  const float* bn2b = (const float*)d_in[13];
  const float* fc1W = (const float*)d_in[14];
  const float* fc1b = (const float*)d_in[15];
  const float* fc2W = (const float*)d_in[16];
  const float* fc2b = (const float*)d_in[17];
  float* out = (float*)d_out;

  // workspace layout (floats); buffers ordered so padded-tile overreads stay in-ws
  float* w = (float*)d_ws;
  const long szZ  = (long)T_ * DZ;           // 15,513,600
  const long szH  = (long)T_ * DZ;           // per-head Q/K/V half
  const long szSC = (long)2 * B_ * L_ * SCP; // 10,756,096
  const long szM  = (long)T_ * DM;           // 31,027,200
  float* z    = w;
  float* Q    = z   + szZ;
  float* K    = Q   + 2 * szH;
  float* V    = K   + 2 * szH;
  float* sc   = V   + 2 * szH;
  float* msa  = sc  + szSC;
  float* y1   = msa + szM;
  float* st   = y1  + szZ;                   // 512 floats of stats
  float* mean1 = st, *rstd1 = st + 128, *mean2 = st + 256, *rstd2 = st + 384;
  float* hid  = st + 512;                    // 512*512

  // 1) assemble z (cls row + positional halves)
  {
    long total = (long)B_ * L_ * DE + (long)B_ * DE;
    k_fill_z<<<dim3((unsigned)((total + 255) / 256)), 256, 0, stream>>>(z, pos, cls);
  }
  // 2) patch embed -> z[:, :150]
  k_gemm_patch<<<dim3(3, 3200), 32, 0, stream>>>(x, Wp, bp, z);
  // 3) Q/K/V projections, both heads
  for (int h = 0; h < 2; ++h) {
    k_gemm_proj<<<dim3(5, 3232), 32, 0, stream>>>(z, Wq + (long)h * DZ * DZ, Q + (long)h * szH);
    k_gemm_proj<<<dim3(5, 3232), 32, 0, stream>>>(z, Wk + (long)h * DZ * DZ, K + (long)h * szH);
    k_gemm_proj<<<dim3(5, 3232), 32, 0, stream>>>(z, Wv + (long)h * DZ * DZ, V + (long)h * szH);
  }
  // 4) attention scores + softmax + attn@V
  k_gemm_scores<<<dim3(2, 7, 2 * B_), 32, 0, stream>>>(Q, K, sc);
  k_softmax<<<dim3(2 * B_ * L_), 32, 0, stream>>>(sc);
  k_gemm_attnv<<<dim3(5, 7, 2 * B_), 32, 0, stream>>>(sc, V, msa);
  // 5) BN1 on msa (in-place)
  k_bn_stats<<<dim3(L_), 256, 0, stream>>>(msa, DM, mean1, rstd1);
  {
    long total = szM;
    k_bn_apply<<<dim3((unsigned)((total + 255) / 256)), 256, 0, stream>>>(msa, DM, mean1, rstd1, bn1g, bn1b, total);
  }
  // 6) y1 = relu(concat(msa_bn, z) @ Wp1 + bp1)
  k_gemm_p1<<<dim3(5, 3232), 32, 0, stream>>>(msa, z, Wp1, bp1, y1);
  // 7) BN2 on y1 (in-place)
  k_bn_stats<<<dim3(L_), 256, 0, stream>>>(y1, DZ, mean2, rstd2);
  {
    long total = szZ;
    k_bn_apply<<<dim3((unsigned)((total + 255) / 256)), 256, 0, stream>>>(y1, DZ, mean2, rstd2, bn2g, bn2b, total);
  }
  // 8) heads: fc1 (WMMA, 512x1200x512) then fc2 (tiny)
  k_gemm_fc1<<<dim3(8, 32), 32, 0, stream>>>(y1, msa, z, fc1W, fc1b, hid);
  k_fc2<<<dim3(4), 256, 0, stream>>>(hid, fc2W, fc2b, out);
  (void)in_sizes; (void)n_in; (void)out_size; (void)ws_size;
}